// GCN_6846177870285
// MI455X (gfx1250) — compile-verified
//
#include <hip/hip_runtime.h>

typedef __attribute__((ext_vector_type(2))) float v2f;
typedef __attribute__((ext_vector_type(8))) float v8f;

#define F_OUT 64

// ---------------- utility kernels ----------------

__global__ void fill_f32(float* __restrict__ p, float v, long long n) {
    long long t = (long long)blockIdx.x * blockDim.x + threadIdx.x;
    if (t < n) p[t] = v;
}

__global__ void count_deg(const long long* __restrict__ dst, float* __restrict__ deg, int E) {
    int e = blockIdx.x * blockDim.x + threadIdx.x;
    if (e < E) atomicAdd(&deg[(int)dst[e]], 1.0f);
}

__global__ void rsqrt_inplace(float* __restrict__ p, int n) {
    int t = blockIdx.x * blockDim.x + threadIdx.x;
    if (t < n) p[t] = rsqrtf(p[t]);   // deg >= 1 always (self-loop), matches reference
}

// ---------------- WMMA GEMM: C[N,64] = A[N,K] * B[K,64] ----------------
// Block = 128 threads = 4 waves; wave w computes 16x16 tile at cols [16w,16w+16).
// f32 WMMA 16x16x4: A VGPR0/1 = K {k0+2hi, k0+2hi+1} for lane%16 = M row;
// B mirrored (rows of B striped across lanes); C VGPR v = row (v + 8*hi).
template <int K>
__global__ __launch_bounds__(128) void gemm_wmma_f32(const float* __restrict__ A,
                                                     const float* __restrict__ B,
                                                     float* __restrict__ C, int N) {
    const int lane = threadIdx.x & 31;
    const int wave = threadIdx.x >> 5;
    const int hi = lane >> 4;
    const int lm = lane & 15;
    const int rowbase = blockIdx.x * 16;
    int arow = rowbase + lm;
    if (arow >= N) arow = N - 1;           // clamp OOB reads; stores are guarded
    const int col = wave * 16 + lm;

    const float* __restrict__ ap = A + (size_t)arow * K;
    v8f acc = {};
#pragma unroll
    for (int kk = 0; kk < K; kk += 4) {
        const int k0 = kk + 2 * hi;
        float2 av = *(const float2*)(ap + k0);
        v2f a; a.x = av.x; a.y = av.y;
        v2f b; b.x = B[k0 * F_OUT + col]; b.y = B[(k0 + 1) * F_OUT + col];
        acc = __builtin_amdgcn_wmma_f32_16x16x4_f32(
            /*neg_a=*/false, a, /*neg_b=*/false, b,
            /*c_mod=*/(short)0, acc, /*reuse_a=*/false, /*reuse_b=*/false);
    }
#pragma unroll
    for (int v = 0; v < 8; ++v) {
        const int r = rowbase + v + 8 * hi;
        if (r < N) C[(size_t)r * F_OUT + col] = acc[v];
    }
}

// ---------------- edge aggregation: agg[dst] += norm * h[src] ----------------
// 16 lanes per edge, 4 features per lane (64 features). h stays L2-resident.
__global__ void edge_agg(const long long* __restrict__ src, const long long* __restrict__ dst,
                         const float* __restrict__ dis, const float* __restrict__ h,
                         float* __restrict__ agg, int E) {
    long long t = (long long)blockIdx.x * blockDim.x + threadIdx.x;
    int e = (int)(t >> 4);
    if (e >= E) return;
    int f = ((int)t & 15) << 2;
    int s = (int)src[e];
    int d = (int)dst[e];
    float nrm = dis[s] * dis[d];
    float4 hv = *(const float4*)(h + (size_t)s * F_OUT + f);
    float* o = agg + (size_t)d * F_OUT + f;
    atomicAdd(o + 0, hv.x * nrm);
    atomicAdd(o + 1, hv.y * nrm);
    atomicAdd(o + 2, hv.z * nrm);
    atomicAdd(o + 3, hv.w * nrm);
}

// ---------------- epilogues (fold self-loop: + dis[i]^2 * h[i]) ----------------

__global__ void relu_combine(float* __restrict__ h, const float* __restrict__ agg,
                             const float* __restrict__ dis, const float* __restrict__ bias,
                             int total) {
    int t = blockIdx.x * blockDim.x + threadIdx.x;
    if (t >= total) return;
    int i = t >> 6, f = t & 63;
    float d = dis[i];
    float v = agg[t] + d * d * h[t] + bias[f];
    h[t] = fmaxf(v, 0.0f);
}

__global__ void final_combine(float* __restrict__ out, const float* __restrict__ h,
                              const float* __restrict__ dis, const float* __restrict__ bias,
                              int total) {
    int t = blockIdx.x * blockDim.x + threadIdx.x;
    if (t >= total) return;
    int i = t >> 6, f = t & 63;
    float d = dis[i];
    out[t] = out[t] + d * d * h[t] + bias[f];
}

// ---------------- host launcher ----------------

extern "C" void kernel_launch(void* const* d_in, const int* in_sizes, int n_in,
                              void* d_out, int out_size, void* d_ws, size_t ws_size,
                              hipStream_t stream) {
    const float*     x  = (const float*)d_in[0];
    const long long* ei = (const long long*)d_in[1];
    const float*     W1 = (const float*)d_in[2];
    const float*     b1 = (const float*)d_in[3];
    const float*     W2 = (const float*)d_in[4];
    const float*     b2 = (const float*)d_in[5];

    const int N = in_sizes[0] / 128;     // F_in = 128
    const int E = in_sizes[1] / 2;       // edge_index is [2, E]
    const long long* src = ei;
    const long long* dst = ei + E;

    float* out  = (float*)d_out;
    float* dis  = (float*)d_ws;                       // [N]     degree -> dis in place
    float* bufA = dis + N;                            // [N*64]  h1 -> relu(h1)
    float* bufB = bufA + (size_t)N * F_OUT;           // [N*64]  agg1 -> h2

    const long long NF = (long long)N * F_OUT;
    const int TB = 256;

    // --- zero/init accumulators (every call; ws/out are not re-poisoned) ---
    fill_f32<<<(int)((N + TB - 1) / TB), TB, 0, stream>>>(dis, 1.0f, N);      // self-loop deg
    fill_f32<<<(int)((NF + TB - 1) / TB), TB, 0, stream>>>(bufB, 0.0f, NF);   // agg1
    fill_f32<<<(int)((NF + TB - 1) / TB), TB, 0, stream>>>(out, 0.0f, NF);    // agg2 -> out

    // --- degree + normalization ---
    count_deg<<<(E + TB - 1) / TB, TB, 0, stream>>>(dst, dis, E);
    rsqrt_inplace<<<(N + TB - 1) / TB, TB, 0, stream>>>(dis, N);

    // --- layer 1 ---
    const int gemm_grid = (N + 15) / 16;
    gemm_wmma_f32<128><<<gemm_grid, 128, 0, stream>>>(x, W1, bufA, N);        // h1 = x @ W1
    {
        long long work = (long long)E * 16;
        edge_agg<<<(int)((work + TB - 1) / TB), TB, 0, stream>>>(src, dst, dis, bufA, bufB, E);
    }
    relu_combine<<<(int)((NF + TB - 1) / TB), TB, 0, stream>>>(bufA, bufB, dis, b1, (int)NF);

    // --- layer 2 ---
    gemm_wmma_f32<64><<<gemm_grid, 128, 0, stream>>>(bufA, W2, bufB, N);      // h2 = h1 @ W2
    {
        long long work = (long long)E * 16;
        edge_agg<<<(int)((work + TB - 1) / TB), TB, 0, stream>>>(src, dst, dis, bufB, out, E);
    }
    final_combine<<<(int)((NF + TB - 1) / TB), TB, 0, stream>>>(out, bufB, dis, b2, (int)NF);
}